// HetSANN_87514253623553
// MI455X (gfx1250) — compile-verified
//
#include <hip/hip_runtime.h>

#define N_NODES 50000
#define N_EDGES 800000
#define N_ETYPES 5
#define DIM 64
#define NEG_SLOPE 0.2f
#define NTILES (N_NODES / 16)          // 3125 exactly

typedef float v2f __attribute__((ext_vector_type(2)));
typedef float v8f __attribute__((ext_vector_type(8)));

// ---------------------------------------------------------------------------
// Precompute fused, transposed weights for both layers.
// wcatT[t][c][k], c in [0,80):
//   layer0: c<64 -> W0[t][k][c]; c in [64,72) -> (W0[t] @ al0[t])[k][c-64]
//           c in [72,80) -> (W0[t] @ ar0[t])[k][c-72]
//           al0[t][j][i] = sum_{d<8} Al0[t][j][i*8+d]   (per-head column sum)
//   layer1: c<64 -> W1[t][k][c]; c==64 -> W1[t]@al1vec; c==65 -> W1[t]@ar1vec;
//           c>=66 -> 0   (al1vec[j] = sum over all 64 cols of Al1[t][j][:])
// ---------------------------------------------------------------------------
__global__ void hs_precompute(const float* __restrict__ W0, const float* __restrict__ Al0,
                              const float* __restrict__ Ar0,
                              const float* __restrict__ W1, const float* __restrict__ Al1,
                              const float* __restrict__ Ar1,
                              float* __restrict__ wcat0, float* __restrict__ wcat1) {
    int idx = blockIdx.x * blockDim.x + threadIdx.x;
    if (idx >= N_ETYPES * 80 * DIM) return;
    int k = idx % DIM;
    int c = (idx / DIM) % 80;
    int t = idx / (DIM * 80);
    const float* W0t = W0 + (size_t)t * DIM * DIM;
    const float* W1t = W1 + (size_t)t * DIM * DIM;

    float v0;
    if (c < 64) {
        v0 = W0t[k * DIM + c];
    } else {
        int head = (c < 72) ? (c - 64) : (c - 72);
        const float* A = (c < 72) ? Al0 : Ar0;
        float s = 0.f;
        for (int j = 0; j < DIM; ++j) {
            const float* Arow = A + ((size_t)t * DIM + j) * DIM + head * 8;
            float a = 0.f;
            for (int d = 0; d < 8; ++d) a += Arow[d];
            s += W0t[k * DIM + j] * a;
        }
        v0 = s;
    }
    wcat0[((size_t)t * 80 + c) * DIM + k] = v0;

    float v1 = 0.f;
    if (c < 64) {
        v1 = W1t[k * DIM + c];
    } else if (c == 64 || c == 65) {
        const float* A = (c == 64) ? Al1 : Ar1;
        float s = 0.f;
        for (int j = 0; j < DIM; ++j) {
            const float* Arow = A + ((size_t)t * DIM + j) * DIM;
            float a = 0.f;
            for (int d = 0; d < DIM; ++d) a += Arow[d];
            s += W1t[k * DIM + j] * a;
        }
        v1 = s;
    }
    wcat1[((size_t)t * 80 + c) * DIM + k] = v1;
}

// ---------------------------------------------------------------------------
// Per-(node-tile-of-16, etype) fused GEMM using V_WMMA_F32_16X16X4_F32.
// One wave handles M=16 nodes x N=80 cols (5 WMMA tiles), K=64 (16 k-steps).
// Tiles 0..3 -> X[t][node][0..63]; tile 4 -> HL(heads cols)/HR -> attention.
//   A layout: lane l holds M=l%16, K = r + 2*(l>=16)  -> float2 @ feat[m][k0+2h]
//   B layout: lane l holds N=l%16, K = r + 2*(l>=16)  -> float2 @ wcatT[c][k0+2h]
//   C/D     : vgpr v, lane l -> M = v + 8*(l>=16), N = l%16
// ---------------------------------------------------------------------------
__global__ __launch_bounds__(256)
void hs_node_gemm_wmma(const float* __restrict__ feat,   // [N,64]
                       const float* __restrict__ wcatT,  // [T][80][64]
                       float* __restrict__ X,            // [T][N][64]
                       float* __restrict__ att,          // heads=8: [T][N][8]; heads=1: [T][N]
                       int heads) {
    int wave = (blockIdx.x * blockDim.x + threadIdx.x) >> 5;
    int lane = threadIdx.x & 31;
    if (wave >= NTILES * N_ETYPES) return;       // wave-uniform exit: EXEC stays all-1s
    int t    = wave % N_ETYPES;
    int tile = wave / N_ETYPES;
    int nl = lane & 15;
    int hi = lane >> 4;

    const float* arow  = feat + ((size_t)(tile * 16 + nl)) * DIM;
    const float* bbase = wcatT + (size_t)t * 80 * DIM;

    v8f acc[5];
#pragma unroll
    for (int j = 0; j < 5; ++j)
#pragma unroll
        for (int v = 0; v < 8; ++v) acc[j][v] = 0.f;

#pragma unroll
    for (int kk = 0; kk < 16; ++kk) {
        int k0 = kk * 4 + hi * 2;
        v2f a = *(const v2f*)(arow + k0);
#pragma unroll
        for (int j = 0; j < 5; ++j) {
            v2f b = *(const v2f*)(bbase + (size_t)(j * 16 + nl) * DIM + k0);
            acc[j] = __builtin_amdgcn_wmma_f32_16x16x4_f32(
                false, a, false, b, (short)0, acc[j], false, false);
        }
    }

    int node0 = tile * 16 + hi * 8;   // row for vgpr v is node0 + v
#pragma unroll
    for (int j = 0; j < 4; ++j)
#pragma unroll
        for (int v = 0; v < 8; ++v)
            X[((size_t)t * N_NODES + node0 + v) * DIM + j * 16 + nl] = acc[j][v];

    // attention from tile 4: HL in cols 0..heads-1, HR in cols heads..2*heads-1
    if (heads == 8) {
#pragma unroll
        for (int v = 0; v < 8; ++v) {
            float x = acc[4][v];
            float other = __shfl_xor(x, 8, 32);
            if (nl < 8) {
                float s  = x + other;
                float lr = s > 0.f ? s : NEG_SLOPE * s;
                att[((size_t)t * N_NODES + node0 + v) * 8 + nl] =
                    1.f / (1.f + __expf(-lr));
            }
        }
    } else {
#pragma unroll
        for (int v = 0; v < 8; ++v) {
            float x = acc[4][v];
            float other = __shfl_xor(x, 1, 32);
            if (nl == 0) {
                float s  = x + other;
                float lr = s > 0.f ? s : NEG_SLOPE * s;
                att[(size_t)t * N_NODES + node0 + v] = 1.f / (1.f + __expf(-lr));
            }
        }
    }
}

// ---------------------------------------------------------------------------
// Edge scatter, layer 0 (8 heads). out[dst][d*8+h] += X[t][src][h*8+d]*att[h]*w
// One wave per edge-group of 4; each lane covers output elements lane, lane+32.
// All hot data (X 64MB, out 12.8MB) is L2-resident; atomics resolve in L2.
// ---------------------------------------------------------------------------
__global__ __launch_bounds__(256)
void hs_edge_scatter_l0(const int* __restrict__ src, const int* __restrict__ dst,
                        const int* __restrict__ etype, const float* __restrict__ ew,
                        const float* __restrict__ X, const float* __restrict__ att,
                        float* __restrict__ out) {
    int wave = (blockIdx.x * blockDim.x + threadIdx.x) >> 5;
    int lane = threadIdx.x & 31;
    long e0 = (long)wave * 4;
    for (int i = 0; i < 4; ++i) {
        long e = e0 + i;
        if (e >= N_EDGES) return;
        int s = src[e], d = dst[e], t = etype[e];
        float w = ew[e];
        const float* xr = X + ((size_t)t * N_NODES + s) * DIM;
        const float* ar = att + ((size_t)t * N_NODES + s) * 8;
        float* orow = out + (size_t)d * DIM;
#pragma unroll
        for (int rep = 0; rep < 2; ++rep) {
            int j  = lane + rep * 32;      // j = dd*8 + hh
            int dd = j >> 3, hh = j & 7;
            atomicAdd(orow + j, xr[hh * 8 + dd] * ar[hh] * w);
        }
    }
}

// Edge scatter, layer 1 (1 head): out[dst][j] += X[t][src][j] * att[t][src] * w
__global__ __launch_bounds__(256)
void hs_edge_scatter_l1(const int* __restrict__ src, const int* __restrict__ dst,
                        const int* __restrict__ etype, const float* __restrict__ ew,
                        const float* __restrict__ X, const float* __restrict__ att,
                        float* __restrict__ out) {
    int wave = (blockIdx.x * blockDim.x + threadIdx.x) >> 5;
    int lane = threadIdx.x & 31;
    long e0 = (long)wave * 4;
    for (int i = 0; i < 4; ++i) {
        long e = e0 + i;
        if (e >= N_EDGES) return;
        int s = src[e], d = dst[e], t = etype[e];
        float a = att[(size_t)t * N_NODES + s] * ew[e];
        const float* xr = X + ((size_t)t * N_NODES + s) * DIM;
        float* orow = out + (size_t)d * DIM;
#pragma unroll
        for (int rep = 0; rep < 2; ++rep) {
            int j = lane + rep * 32;
            atomicAdd(orow + j, xr[j] * a);
        }
    }
}

__global__ void hs_zero(float* __restrict__ p, int n) {
    int i = blockIdx.x * blockDim.x + threadIdx.x;
    if (i < n) p[i] = 0.f;
}

__global__ void hs_elu(const float* __restrict__ in, float* __restrict__ outp, int n) {
    int i = blockIdx.x * blockDim.x + threadIdx.x;
    if (i < n) {
        float x = in[i];
        outp[i] = x > 0.f ? x : (__expf(x) - 1.f);
    }
}

// d_out = h1 @ Wres + bres  (initializes output; edge scatter L1 adds on top)
__global__ void hs_residual(const float* __restrict__ h1, const float* __restrict__ Wres,
                            const float* __restrict__ bres, float* __restrict__ out) {
    int idx = blockIdx.x * blockDim.x + threadIdx.x;
    if (idx >= N_NODES * DIM) return;
    int c = idx & 63;
    int n = idx >> 6;
    const float* hr = h1 + (size_t)n * DIM;
    float s = bres[c];
#pragma unroll 8
    for (int k = 0; k < DIM; ++k) s += hr[k] * Wres[k * DIM + c];
    out[idx] = s;
}

extern "C" void kernel_launch(void* const* d_in, const int* in_sizes, int n_in,
                              void* d_out, int out_size, void* d_ws, size_t ws_size,
                              hipStream_t stream) {
    const float* feat  = (const float*)d_in[0];
    const float* ew    = (const float*)d_in[1];
    const int*   src   = (const int*)d_in[2];
    const int*   dst   = (const int*)d_in[3];
    const int*   etype = (const int*)d_in[4];
    const float* W0    = (const float*)d_in[5];
    const float* Al0   = (const float*)d_in[6];
    const float* Ar0   = (const float*)d_in[7];
    const float* W1    = (const float*)d_in[8];
    const float* Al1   = (const float*)d_in[9];
    const float* Ar1   = (const float*)d_in[10];
    const float* Wres  = (const float*)d_in[11];
    const float* bres  = (const float*)d_in[12];
    float* out = (float*)d_out;

    char* ws = (char*)d_ws;
    size_t off = 0;
    auto alloc = [&](size_t bytes) -> void* {
        off = (off + 255) & ~(size_t)255;
        void* p = ws + off;
        off += bytes;
        return p;
    };
    float* wcat0 = (float*)alloc((size_t)N_ETYPES * 80 * DIM * 4);
    float* wcat1 = (float*)alloc((size_t)N_ETYPES * 80 * DIM * 4);
    float* X     = (float*)alloc((size_t)N_ETYPES * N_NODES * DIM * 4);  // reused by both layers
    float* att   = (float*)alloc((size_t)N_ETYPES * N_NODES * 8 * 4);    // layer1 reuses ([T][N])
    float* out0  = (float*)alloc((size_t)N_NODES * DIM * 4);
    float* h1    = (float*)alloc((size_t)N_NODES * DIM * 4);

    const int B = 256;
    int gPre  = (N_ETYPES * 80 * DIM + B - 1) / B;
    int gNode = (N_NODES * DIM + B - 1) / B;
    int gGemm = (NTILES * N_ETYPES + 7) / 8;               // 8 waves/block
    int gEdge = ((N_EDGES + 3) / 4 + 7) / 8;               // 4 edges/wave, 8 waves/block

    hs_precompute<<<gPre, B, 0, stream>>>(W0, Al0, Ar0, W1, Al1, Ar1, wcat0, wcat1);
    hs_zero<<<gNode, B, 0, stream>>>(out0, N_NODES * DIM);

    // ---- layer 0 ----
    hs_node_gemm_wmma<<<gGemm, B, 0, stream>>>(feat, wcat0, X, att, 8);
    hs_edge_scatter_l0<<<gEdge, B, 0, stream>>>(src, dst, etype, ew, X, att, out0);
    hs_elu<<<gNode, B, 0, stream>>>(out0, h1, N_NODES * DIM);

    // ---- layer 1 ----
    hs_node_gemm_wmma<<<gGemm, B, 0, stream>>>(h1, wcat1, X, att, 1);
    hs_residual<<<gNode, B, 0, stream>>>(h1, Wres, bres, out);
    hs_edge_scatter_l1<<<gEdge, B, 0, stream>>>(src, dst, etype, ew, X, att, out);
}